// PLabelingNet_41351945126301
// MI455X (gfx1250) — compile-verified
//
#include <hip/hip_runtime.h>

#define N_NODES 10000
#define N_EDGES 160000
#define BSZ 16
#define FEAT 64

typedef __attribute__((ext_vector_type(2))) float v2f;
typedef __attribute__((ext_vector_type(8))) float v8f;

#if defined(__has_builtin)
#if __has_builtin(__builtin_amdgcn_global_load_async_to_lds_b128)
#define HAVE_ASYNC_LDS 1
#endif
#endif

#ifdef HAVE_ASYNC_LDS
// builtin expects: (int4vec addrspace(1)*, int4vec addrspace(3)*, imm offset, imm cpol)
typedef int v4i_vs __attribute__((vector_size(16)));
typedef __attribute__((address_space(1))) v4i_vs GV4;
typedef __attribute__((address_space(3))) v4i_vs LV4;
#endif

__device__ __forceinline__ void wait_async_lds() {
#if defined(__has_builtin) && __has_builtin(__builtin_amdgcn_s_wait_asynccnt)
    __builtin_amdgcn_s_wait_asynccnt(0);
#else
    asm volatile("s_wait_asynccnt 0x0" ::: "memory");
#endif
}

// ---------------------------------------------------------------- zero scratch
__global__ void k_zero(float* __restrict__ p, int n) {
    for (int i = blockIdx.x * blockDim.x + threadIdx.x; i < n; i += gridDim.x * blockDim.x)
        p[i] = 0.0f;
}

// ------------------------------------------------- unique(pos) / inverse / targets
// pos has 16 distinct node ids; rank = #smaller gives the sorted-unique index.
// Target t = 2m+c: out[m][c] = h_final[inv[m][1-c], pos[m][c]]  -> tgt_b[t] = rank[t^1].
__global__ void k_pos(const int* __restrict__ pos, int* __restrict__ sel_id,
                      int* __restrict__ tgt_b, int* __restrict__ tgt_n) {
    __shared__ int sp[BSZ], sr[BSZ];
    int i = threadIdx.x;
    if (i < BSZ) sp[i] = pos[i];
    __syncthreads();
    if (i < BSZ) {
        int r = 0;
        for (int j = 0; j < BSZ; ++j) r += (sp[j] < sp[i]) ? 1 : 0;
        sr[i] = r;
        sel_id[r] = sp[i];
    }
    __syncthreads();
    if (i < BSZ) { tgt_n[i] = sp[i]; tgt_b[i] = sr[i ^ 1]; }
}

// ------------------------------------------------- WMMA f32 GEMM: C = act(A) @ W + bias
// A: (nrows,64) f32 row-major (16B aligned), W: (64,64) row-major, bias: (64)
// Requires nrows % 16 == 0 (all call sites use nrows = 10000).
// Block: 256 thr = 8 waves, covers 32 rows x 64 cols.
// Wave w: row-tile (w>>2)*16, col-tile (w&3)*16. 16 k-steps of 16x16x4 f32 WMMA.
// A tile staged into LDS via gfx1250 async-to-LDS (ASYNCcnt) when available.
// ACT is a compile-time flag so the non-ReLU instances carry zero VALU overhead.
template <int ACT>
__global__ __launch_bounds__(256) void k_gemm64(const float* __restrict__ A,
                                                const float* __restrict__ Wm,
                                                const float* __restrict__ bias,
                                                float* __restrict__ C,
                                                int nrows) {
    __shared__ __align__(16) float As[32 * FEAT];   // 8 KB contiguous A tile
    const int tid = threadIdx.x;
    const int r0blk = blockIdx.x * 32;

    // ---- stage A tile: 512 x 16B chunks (chunk i: row = i>>4, col4 = i&15)
#ifdef HAVE_ASYNC_LDS
    for (int i = tid; i < 512; i += 256) {
        const int row = r0blk + (i >> 4);
        if (row < nrows) {   // EXEC-masked lanes skip; OOB rows never read back
            __builtin_amdgcn_global_load_async_to_lds_b128(
                (GV4*)(A + row * FEAT + ((i & 15) << 2)),
                (LV4*)&As[i << 2], 0, 0);
        }
    }
    wait_async_lds();
#else
    const float4* __restrict__ A4 = (const float4*)A;
    float4* __restrict__ As4 = (float4*)As;
    for (int i = tid; i < 512; i += 256) {
        const int row = r0blk + (i >> 4);
        float4 v = make_float4(0.f, 0.f, 0.f, 0.f);
        if (row < nrows) v = A4[row * 16 + (i & 15)];
        As4[i] = v;
    }
#endif
    __syncthreads();

    const int wave = tid >> 5;
    const int lane = tid & 31;
    const int rtile = (wave >> 2) * 16;        // 0 or 16 inside block
    const int c0 = (wave & 3) * 16;            // column block
    const int r0 = r0blk + rtile;
    if (r0 >= nrows) return;                   // wave-uniform: EXEC stays all-ones for WMMA

    const int mrow = lane & 15;                // M (for A) / N (for B) index
    const int koff = (lane >> 4) << 1;         // lanes 16-31 hold K+2,K+3

    const float bv = bias[c0 + mrow];
    v8f c = { bv, bv, bv, bv, bv, bv, bv, bv };

#pragma unroll
    for (int t = 0; t < 16; ++t) {
        const int k = 4 * t + koff;
        v2f a, bf;
        // A 16x4 f32 layout: lane<16 -> K=k0,k0+1 ; lane>=16 -> K=k0+2,k0+3
        a.x = As[(rtile + mrow) * FEAT + k];
        a.y = As[(rtile + mrow) * FEAT + k + 1];
        if (ACT) {  // plain select ReLU: cmp + cndmask, dual-issuable
            a.x = (a.x > 0.0f) ? a.x : 0.0f;
            a.y = (a.y > 0.0f) ? a.y : 0.0f;
        }
        // B 4x16 f32 layout (rows striped across lanes within a VGPR)
        bf.x = Wm[k * FEAT + c0 + mrow];
        bf.y = Wm[(k + 1) * FEAT + c0 + mrow];
        c = __builtin_amdgcn_wmma_f32_16x16x4_f32(false, a, false, bf,
                                                  (short)0, c, false, false);
    }
    // C/D 16x16 f32 layout: VGPR v: lanes 0-15 -> M=v, lanes 16-31 -> M=v+8.
    // Tile fully in-bounds (nrows % 16 == 0): unguarded store clause.
    const int rbase = r0 + ((lane >> 4) << 3);
#pragma unroll
    for (int v = 0; v < 8; ++v) {
        C[(rbase + v) * FEAT + c0 + mrow] = c[v];
    }
}

// ------------------------------------------------- layer-0 small: Delta0 -> G0
// x1_0[b] = x[sel]@f1w0+f1b0 ; Delta0 = x1_0 - base[sel] ; G0 = Delta0 @ conv_w0
__global__ __launch_bounds__(64) void k_small0(const float* __restrict__ x,
                                               const float* __restrict__ base,
                                               const float* __restrict__ f1w0,
                                               const float* __restrict__ f1b0,
                                               const float* __restrict__ convw0,
                                               const int* __restrict__ sel_id,
                                               float* __restrict__ G0) {
    __shared__ float xv[FEAT], dv[FEAT];
    const int b = blockIdx.x, j = threadIdx.x;
    const int sel = sel_id[b];
    xv[j] = x[sel * FEAT + j];
    __syncthreads();
    float a = f1b0[j];
    for (int i = 0; i < FEAT; ++i) a += xv[i] * f1w0[i * FEAT + j];
    dv[j] = a - base[sel * FEAT + j];
    __syncthreads();
    float g = 0.0f;
    for (int i = 0; i < FEAT; ++i) g += dv[i] * convw0[i * FEAT + j];
    G0[b * FEAT + j] = g;
}

// ------------------------------------------------- shared edge aggregation + s[b,n]
// aggbase[n] += w_e * base[src_e]   (all edges, batch-independent)
// sArr[b,n]  += w_e  where src_e == sel_id[b]
__global__ __launch_bounds__(256) void k_edge0(const int* __restrict__ esrc,
                                               const int* __restrict__ edst,
                                               const float* __restrict__ ew,
                                               const float* __restrict__ base,
                                               const int* __restrict__ sel_id,
                                               float* __restrict__ aggbase,
                                               float* __restrict__ sArr) {
    __shared__ int selL[BSZ];
    const int tx = threadIdx.x, ty = threadIdx.y;
    if (ty == 0 && tx < BSZ) selL[tx] = sel_id[tx];
    __syncthreads();
    const int estride = gridDim.x * 4;
    for (int e = blockIdx.x * 4 + ty; e < N_EDGES; e += estride) {
        if (tx == 0 && e + estride < N_EDGES) {  // stream-ahead hints -> global_prefetch_b8
            __builtin_prefetch(&esrc[e + estride], 0, 1);
            __builtin_prefetch(&edst[e + estride], 0, 1);
            __builtin_prefetch(&ew[e + estride], 0, 1);
        }
        const int src = esrc[e], dst = edst[e];
        const float w = ew[e];
        atomicAdd(&aggbase[dst * FEAT + tx], w * base[src * FEAT + tx]);
        if (tx < BSZ && selL[tx] == src) atomicAdd(&sArr[tx * N_NODES + dst], w);
    }
}

// ------------------------------------------------- x1_1[b] = h1[b,sel[b]] @ f1w1 + f1b1
__global__ __launch_bounds__(64) void k_small1(const float* __restrict__ Hbase0,
                                               const float* __restrict__ sArr,
                                               const float* __restrict__ G0,
                                               const int* __restrict__ sel_id,
                                               const float* __restrict__ f1w1,
                                               const float* __restrict__ f1b1,
                                               float* __restrict__ x1_1) {
    __shared__ float hv[FEAT];
    const int b = blockIdx.x, j = threadIdx.x;
    const int sel = sel_id[b];
    const float sv = sArr[b * N_NODES + sel];
    const float h = Hbase0[sel * FEAT + j] + sv * G0[b * FEAT + j];
    hv[j] = (h > 0.0f) ? h : 0.0f;
    __syncthreads();
    float a = f1b1[j];
    for (int i = 0; i < FEAT; ++i) a += hv[i] * f1w1[i * FEAT + j];
    x1_1[b * FEAT + j] = a;
}

// ------------------------------------------------- compact edges hitting target nodes
__global__ __launch_bounds__(256) void k_match(const int* __restrict__ edst,
                                               const int* __restrict__ tgt_n,
                                               int* __restrict__ counter,
                                               int* __restrict__ matches) {
    __shared__ int tn[BSZ];
    if (threadIdx.x < BSZ) tn[threadIdx.x] = tgt_n[threadIdx.x];
    __syncthreads();
    for (int e = blockIdx.x * blockDim.x + threadIdx.x; e < N_EDGES;
         e += gridDim.x * blockDim.x) {
        const int d = edst[e];
#pragma unroll
        for (int t = 0; t < BSZ; ++t)
            if (tn[t] == d) matches[atomicAdd(counter, 1)] = (t << 24) | e;
    }
}

// ------------------------------------------------- accumulate agg2 at the 16 targets
// acc[t] += w_e * x0_1[b_t, src_e]; x0_1 = x1_1 (override) / corrected / S1 (shared)
__global__ __launch_bounds__(64) void k_process(const int* __restrict__ counter,
                                                const int* __restrict__ matches,
                                                const int* __restrict__ tgt_b,
                                                const int* __restrict__ sel_id,
                                                const int* __restrict__ esrc,
                                                const float* __restrict__ ew,
                                                const float* __restrict__ sArr,
                                                const float* __restrict__ Hbase0,
                                                const float* __restrict__ G0,
                                                const float* __restrict__ S1,
                                                const float* __restrict__ x1_1,
                                                const float* __restrict__ f0w1,
                                                const float* __restrict__ f0b1,
                                                float* __restrict__ accBuf) {
    __shared__ float hv[FEAT];
    const int j = threadIdx.x;
    const int cnt = counter[0];
    for (int idx = blockIdx.x; idx < cnt; idx += gridDim.x) {
        const int pk = matches[idx];
        const int t = pk >> 24;
        const int e = pk & 0xFFFFFF;
        const int b = tgt_b[t];
        const int src = esrc[e];
        const float w = ew[e];
        float vj;
        if (src == sel_id[b]) {                 // block-uniform branch
            vj = x1_1[b * FEAT + j];
        } else {
            const float sv = sArr[b * N_NODES + src];   // same value for all j
            if (sv != 0.0f) {                   // corrected node: recompute on the fly
                const float h = Hbase0[src * FEAT + j] + sv * G0[b * FEAT + j];
                hv[j] = (h > 0.0f) ? h : 0.0f;
                __syncthreads();
                float a = f0b1[j];
                for (int i = 0; i < FEAT; ++i) a += hv[i] * f0w1[i * FEAT + j];
                __syncthreads();
                vj = a;
            } else {
                vj = S1[src * FEAT + j];
            }
        }
        atomicAdd(&accBuf[t * FEAT + j], w * vj);
    }
}

// ------------------------------------------------- out[t] = acc[t] @ conv_w1 + conv_b1
__global__ __launch_bounds__(64) void k_finalize(const float* __restrict__ accBuf,
                                                 const float* __restrict__ convw1,
                                                 const float* __restrict__ convb1,
                                                 float* __restrict__ out) {
    __shared__ float av[FEAT];
    const int t = blockIdx.x, j = threadIdx.x;
    av[j] = accBuf[t * FEAT + j];
    __syncthreads();
    float o = convb1[j];
    for (int i = 0; i < FEAT; ++i) o += av[i] * convw1[i * FEAT + j];
    out[t * FEAT + j] = o;   // t = 2m+c -> flat (m,c,j) = t*64+j
}

// ================================================================ launch
extern "C" void kernel_launch(void* const* d_in, const int* in_sizes, int n_in,
                              void* d_out, int out_size, void* d_ws, size_t ws_size,
                              hipStream_t stream) {
    const float* x    = (const float*)d_in[0];
    const int*   pos  = (const int*)d_in[1];
    const int*   esrc = (const int*)d_in[2];
    const int*   edst = (const int*)d_in[3];
    const float* ew   = (const float*)d_in[4];
    const float* f0w  = (const float*)d_in[5];
    const float* f0b  = (const float*)d_in[6];
    const float* f1w  = (const float*)d_in[7];
    const float* f1b  = (const float*)d_in[8];
    const float* cw   = (const float*)d_in[9];
    const float* cb   = (const float*)d_in[10];
    float* out = (float*)d_out;

    // workspace layout (floats)
    float* wsf     = (float*)d_ws;
    float* base    = wsf;                       // N*64
    float* Hbase0  = base    + N_NODES * FEAT;  // N*64
    float* S1      = Hbase0  + N_NODES * FEAT;  // N*64
    float* G0      = S1      + N_NODES * FEAT;  // 16*64
    float* x1_1    = G0      + BSZ * FEAT;      // 16*64
    float* aggbase = x1_1    + BSZ * FEAT;      // N*64  (zeroed)
    float* sArr    = aggbase + N_NODES * FEAT;  // 16*N  (zeroed)
    float* accBuf  = sArr    + BSZ * N_NODES;   // 16*64 (zeroed)
    int*   counter = (int*)(accBuf + BSZ * FEAT);   // 1 (zeroed)
    int*   sel_id  = counter + 1;               // 16
    int*   tgt_b   = sel_id + BSZ;              // 16
    int*   tgt_n   = tgt_b + BSZ;               // 16
    int*   matches = tgt_n + BSZ;               // E

    const int zeroCount = N_NODES * FEAT + BSZ * N_NODES + BSZ * FEAT + 1;
    k_zero<<<512, 256, 0, stream>>>(aggbase, zeroCount);

    k_pos<<<1, 32, 0, stream>>>(pos, sel_id, tgt_b, tgt_n);

    const int gemmBlocks = (N_NODES + 31) / 32;
    // base = x @ f0_w[0] + f0_b[0]
    k_gemm64<0><<<gemmBlocks, 256, 0, stream>>>(x, f0w, f0b, base, N_NODES);

    k_small0<<<BSZ, FEAT, 0, stream>>>(x, base, f1w, f1b, cw, sel_id, G0);

    dim3 eb(64, 4);
    k_edge0<<<512, eb, 0, stream>>>(esrc, edst, ew, base, sel_id, aggbase, sArr);

    // Hbase0 = aggbase @ conv_w[0] + conv_b[0]
    k_gemm64<0><<<gemmBlocks, 256, 0, stream>>>(aggbase, cw, cb, Hbase0, N_NODES);
    // S1 = relu(Hbase0) @ f0_w[1] + f0_b[1]
    k_gemm64<1><<<gemmBlocks, 256, 0, stream>>>(Hbase0, f0w + FEAT * FEAT,
                                                f0b + FEAT, S1, N_NODES);

    k_small1<<<BSZ, FEAT, 0, stream>>>(Hbase0, sArr, G0, sel_id,
                                       f1w + FEAT * FEAT, f1b + FEAT, x1_1);

    k_match<<<256, 256, 0, stream>>>(edst, tgt_n, counter, matches);

    k_process<<<128, 64, 0, stream>>>(counter, matches, tgt_b, sel_id, esrc, ew,
                                      sArr, Hbase0, G0, S1, x1_1,
                                      f0w + FEAT * FEAT, f0b + FEAT, accBuf);

    k_finalize<<<BSZ, FEAT, 0, stream>>>(accBuf, cw + FEAT * FEAT, cb + FEAT, out);
}